// SelfAttention_38208029065764
// MI455X (gfx1250) — compile-verified
//
#include <hip/hip_runtime.h>

typedef float f32x2 __attribute__((ext_vector_type(2)));
typedef float f32x8 __attribute__((ext_vector_type(8)));
typedef unsigned int u32x4 __attribute__((ext_vector_type(4)));
typedef int i32x4 __attribute__((ext_vector_type(4)));
typedef int i32x8 __attribute__((ext_vector_type(8)));

constexpr int BATCH = 32;
constexpr int CH    = 85;     // channels (= attention length)
constexpr int SEQ   = 8192;   // sequence length
constexpr int CP    = 96;     // padded channels (6 x 16)
constexpr int CE    = CH + 1; // 86: extended rows (ones-row trick folds biases in)

// D = A(16x4) * B(4x16) + C, all f32, full precision.
__device__ __forceinline__ f32x8 wmma_f32_16x16x4(f32x2 a, f32x2 b, f32x8 c) {
  return __builtin_amdgcn_wmma_f32_16x16x4_f32(false, a, false, b, (short)0, c,
                                               false, false);
}

// ---------------------------------------------------------------------------
// Tensor Data Mover: async 2-D tile DMA global -> LDS with row padding.
// Descriptor fields per CDNA5 ISA 8.3/8.4: group0 = {count, lds_addr,
// global_addr, type=2}; group1 = {data_size=4B, pad_enable, pad codes,
// tensor dims, tile dims, stride}.  Tracked by TENSORcnt.
// ---------------------------------------------------------------------------
__device__ __forceinline__ void tdm_load_2d(const float* gaddr, unsigned lds_off,
                                            int tile_d0, int tile_d1,
                                            int tensor_d0, int tensor_d1,
                                            long long stride0,
                                            int pad_interval_code,
                                            int pad_amount_code) {
  unsigned long long ga = (unsigned long long)(const void*)gaddr;
  u32x4 g0;
  g0.x = 1u;                                        // count=1 (user descriptor)
  g0.y = lds_off;                                   // LDS byte address
  g0.z = (unsigned)(ga & 0xFFFFFFFFu);              // global_addr[31:0]
  g0.w = (unsigned)((ga >> 32) & 0x01FFFFFFu)       // global_addr[56:32]
         | (2u << 30);                              // type=2 ("image")
  i32x8 g1;
  g1[0] = (2 << 16)                                 // data_size: 4 bytes
        | (1 << 20)                                 // pad_enable
        | (pad_interval_code << 22)                 // pad every 2^(c+1) DWORDs
        | (pad_amount_code << 25);                  // pad (c+1) DWORDs
  g1[1] = (tensor_d0 & 0xFFFF) << 16;               // tensor_dim0[15:0] @bit48
  g1[2] = ((tensor_d0 >> 16) & 0xFFFF)              // tensor_dim0[31:16]
        | ((tensor_d1 & 0xFFFF) << 16);             // tensor_dim1[15:0] @bit80
  g1[3] = ((tensor_d1 >> 16) & 0xFFFF)              // tensor_dim1[31:16]
        | ((tile_d0 & 0xFFFF) << 16);               // tile_dim0 @bit112
  g1[4] = (tile_d1 & 0xFFFF);                       // tile_dim1; tile_dim2=0
  g1[5] = (int)(stride0 & 0xFFFFFFFFll);            // tensor_dim0_stride[31:0]
  g1[6] = (int)((stride0 >> 32) & 0xFFFFll);        // tensor_dim0_stride[47:32]
  g1[7] = 0;                                        // tensor_dim1_stride unused (2D)
  i32x4 z4 = {0, 0, 0, 0};                          // groups 2/3: dims unused (2D)
  i32x8 z8 = {0, 0, 0, 0, 0, 0, 0, 0};
  __builtin_amdgcn_tensor_load_to_lds(g0, g1, z4, z4, z8, 0);
}

__device__ __forceinline__ unsigned lds_offset_of(const void* p) {
  // Generic (flat) LDS addresses carry the wave-relative LDS byte offset in
  // the low 32 bits (aperture in the high bits).
  return (unsigned)(unsigned long long)p;
}

// ---------------------------------------------------------------------------
// Kernel 1: Gram matrix  G[b] = x_ext[b] (96 x L) @ x_ext[b]^T  -> [B,96,96]
// x_ext row c<85 = x[b,c,:], row 85 = 1.0 (bias trick), rows 86..95 = 0.
// Grid (BATCH, 32 chunks of 256). 6 waves; TDM double-buffered 96x64 tiles.
// ---------------------------------------------------------------------------
constexpr int K1_CHUNK = 256;
constexpr int K1_SUB   = 64;
constexpr int K1_PITCH = 68;              // 64-DW rows + 4-DW TDM pad, bank-safe
constexpr int NSUB1    = K1_CHUNK / K1_SUB;
constexpr int NCHUNK1  = SEQ / K1_CHUNK;  // 32

__global__ __launch_bounds__(192) void gram_kernel(const float* __restrict__ x,
                                                   float* __restrict__ G) {
  __shared__ float tile[2][CP * K1_PITCH];  // 2 x 26.1 KB
  const int b    = blockIdx.x;
  const int tid  = threadIdx.x;
  const int wave = tid >> 5;
  const int lane = tid & 31;
  const int ln   = lane & 15;
  const int kh   = lane >> 4;  // 0 -> K{0,1}, 1 -> K{2,3} per ISA 16x4 f32 layout
  const float* xb = x + (size_t)b * CH * SEQ;
  const int l0base = blockIdx.y * K1_CHUNK;

  // Static rows (85 = ones, 86..95 = zeros) written once; TDM only touches
  // rows 0..84, so these survive every subtile.
  for (int idx = tid; idx < 2 * (CP - CH) * K1_SUB; idx += 192) {
    int buf = idx / ((CP - CH) * K1_SUB);
    int r   = idx % ((CP - CH) * K1_SUB);
    int row = CH + r / K1_SUB;
    int col = r % K1_SUB;
    tile[buf][row * K1_PITCH + col] = (row == CH) ? 1.f : 0.f;
  }

  if (wave == 0)
    tdm_load_2d(xb + l0base, lds_offset_of(&tile[0][0]),
                K1_SUB, CH, SEQ, CH, SEQ, /*interval 64DW*/ 5, /*pad 4DW*/ 3);

  f32x8 zero = {0.f, 0.f, 0.f, 0.f, 0.f, 0.f, 0.f, 0.f};
  f32x8 acc[6];
#pragma unroll
  for (int n = 0; n < 6; ++n) acc[n] = zero;

  for (int sub = 0; sub < NSUB1; ++sub) {
    if (wave == 0) __builtin_amdgcn_s_wait_tensorcnt(0);
    __syncthreads();  // current buffer ready; previous compute fully drained
    if (wave == 0 && sub + 1 < NSUB1)
      tdm_load_2d(xb + l0base + (sub + 1) * K1_SUB,
                  lds_offset_of(&tile[(sub + 1) & 1][0]),
                  K1_SUB, CH, SEQ, CH, SEQ, 5, 3);

    const float* t = &tile[sub & 1][0];
#pragma unroll
    for (int k = 0; k < K1_SUB; k += 4) {
      f32x2 a = *(const f32x2*)(t + (wave * 16 + ln) * K1_PITCH + k + 2 * kh);
#pragma unroll
      for (int n = 0; n < 6; ++n) {
        f32x2 bb = *(const f32x2*)(t + (n * 16 + ln) * K1_PITCH + k + 2 * kh);
        acc[n] = wmma_f32_16x16x4(a, bb, acc[n]);
      }
    }
  }

  float* Gb = G + (size_t)b * CP * CP;
#pragma unroll
  for (int n = 0; n < 6; ++n) {
#pragma unroll
    for (int v = 0; v < 8; ++v) {
      int c = wave * 16 + v + 8 * kh;  // C/D layout: vgpr v -> M=v (+8 upper half)
      int d = n * 16 + ln;
      atomicAdd(Gb + c * CP + d, acc[n][v]);
    }
  }
}

// ---------------------------------------------------------------------------
// Kernel 2 (tiny): scores = Wq_ext G Wk_ext^T / sqrt(C); softmax; M_ext = [P Wv | P bv]
// ---------------------------------------------------------------------------
constexpr int P2 = 88;

__global__ __launch_bounds__(256) void attn_kernel(
    const float* __restrict__ G,
    const float* __restrict__ Wq, const float* __restrict__ bq,
    const float* __restrict__ Wk, const float* __restrict__ bk,
    const float* __restrict__ Wv, const float* __restrict__ bv,
    float* __restrict__ Mext) {
  __shared__ float sA[CE * P2];  // holds G, then scores/probs
  __shared__ float sT[CE * P2];  // holds T = G * Wk_ext^T
  const int b = blockIdx.x, tid = threadIdx.x;
  const float* Gb = G + (size_t)b * CP * CP;

  for (int idx = tid; idx < CE * CE; idx += 256) {
    int i = idx / CE, j = idx % CE;
    sA[i * P2 + j] = Gb[i * CP + j];
  }
  __syncthreads();

  // T[i][d] = sum_j G[i][j] * Wk_ext[d][j]   (i in 0..85, d in 0..84)
  for (int idx = tid; idx < CE * CH; idx += 256) {
    int i = idx / CH, d = idx % CH;
    float s = 0.f;
    for (int j = 0; j < CH; ++j) s += sA[i * P2 + j] * Wk[d * CH + j];
    s += sA[i * P2 + CH] * bk[d];
    sT[i * P2 + d] = s;
  }
  __syncthreads();

  // S[c][d] = scale * sum_i Wq_ext[c][i] * T[i][d]  (overwrites sA)
  const float scale = 1.0f / sqrtf((float)CH);
  for (int idx = tid; idx < CH * CH; idx += 256) {
    int c = idx / CH, d = idx % CH;
    float s = 0.f;
    for (int i = 0; i < CH; ++i) s += Wq[c * CH + i] * sT[i * P2 + d];
    s += bq[c] * sT[CH * P2 + d];
    sA[c * P2 + d] = s * scale;
  }
  __syncthreads();

  // Row softmax over d = 0..84 (ATT_LEN == C, truncation is a no-op).
  if (tid < CH) {
    float m = -3.4e38f;
    for (int d = 0; d < CH; ++d) m = fmaxf(m, sA[tid * P2 + d]);
    float sum = 0.f;
    for (int d = 0; d < CH; ++d) {
      float e = expf(sA[tid * P2 + d] - m);
      sA[tid * P2 + d] = e;
      sum += e;
    }
    float r = 1.f / sum;
    for (int d = 0; d < CH; ++d) sA[tid * P2 + d] *= r;
  }
  __syncthreads();

  // M_ext[b] (96x96): cols 0..84 = P@Wv, col 85 = P@bv, rest zero-padded.
  float* Mb = Mext + (size_t)b * CP * CP;
  for (int idx = tid; idx < CP * CP; idx += 256) {
    int c = idx / CP, i = idx % CP;
    float s = 0.f;
    if (c < CH) {
      if (i < CH) {
        for (int d = 0; d < CH; ++d) s += sA[c * P2 + d] * Wv[d * CH + i];
      } else if (i == CH) {
        for (int d = 0; d < CH; ++d) s += sA[c * P2 + d] * bv[d];
      }
    }
    Mb[c * CP + i] = s;
  }
}

// ---------------------------------------------------------------------------
// Kernel 3: out[b] (85 x L) = M_ext[b] (96 x 96) @ x_ext[b] (96 x L)
// Grid (BATCH, 32 chunks of 256). 6 waves; wave w owns a 16-row strip with all
// 24 A-fragments register-resident (reused across 4 subtiles); x tiles are
// TDM double-buffered 96x64 [c][l]; B fragment = (k,k+1) pitch-strided pair
// (ds_load_2addr-able).
// ---------------------------------------------------------------------------
constexpr int K3_CHUNK = 256;
constexpr int K3_SUB   = 64;
constexpr int K3_PITCH = 72;   // 64-DW rows + 8-DW TDM pad; 2*72 % 64 == 16
constexpr int NSUB3    = K3_CHUNK / K3_SUB;

__global__ __launch_bounds__(192) void out_kernel(const float* __restrict__ x,
                                                  const float* __restrict__ Mext,
                                                  float* __restrict__ out) {
  __shared__ float tile[2][CP * K3_PITCH];  // 2 x 27.6 KB, layout [c][l]
  const int b    = blockIdx.x;
  const int tid  = threadIdx.x;
  const int wave = tid >> 5;
  const int lane = tid & 31;
  const int ln   = lane & 15;
  const int kh   = lane >> 4;
  const float* xb = x + (size_t)b * CH * SEQ;
  const int l0base = blockIdx.y * K3_CHUNK;

  // Static x_ext rows: 85 = ones, 86..95 = zeros (TDM writes rows 0..84 only).
  for (int idx = tid; idx < 2 * (CP - CH) * K3_SUB; idx += 192) {
    int buf = idx / ((CP - CH) * K3_SUB);
    int r   = idx % ((CP - CH) * K3_SUB);
    int row = CH + r / K3_SUB;
    int col = r % K3_SUB;
    tile[buf][row * K3_PITCH + col] = (row == CH) ? 1.f : 0.f;
  }

  if (wave == 0)
    tdm_load_2d(xb + l0base, lds_offset_of(&tile[0][0]),
                K3_SUB, CH, SEQ, CH, SEQ, /*interval 64DW*/ 5, /*pad 8DW*/ 7);

  // Preload all 24 A-fragments (M_ext strip rows, contiguous k pairs).
  const float* Ma = Mext + (size_t)b * CP * CP + (wave * 16 + ln) * CP + 2 * kh;
  f32x2 af[24];
#pragma unroll
  for (int kk = 0; kk < 24; ++kk) af[kk] = *(const f32x2*)(Ma + 4 * kk);

  f32x8 zero = {0.f, 0.f, 0.f, 0.f, 0.f, 0.f, 0.f, 0.f};
  float* ob = out + (size_t)b * CH * SEQ + l0base;

  for (int sub = 0; sub < NSUB3; ++sub) {
    if (wave == 0) __builtin_amdgcn_s_wait_tensorcnt(0);
    __syncthreads();  // current buffer ready; previous compute drained
    if (wave == 0 && sub + 1 < NSUB3)
      tdm_load_2d(xb + l0base + (sub + 1) * K3_SUB,
                  lds_offset_of(&tile[(sub + 1) & 1][0]),
                  K3_SUB, CH, SEQ, CH, SEQ, 5, 7);

    const float* t = &tile[sub & 1][0];
    f32x8 acc[4];
#pragma unroll
    for (int n = 0; n < 4; ++n) acc[n] = zero;

#pragma unroll
    for (int kk = 0; kk < 24; ++kk) {
      const float* bp = t + (4 * kk + 2 * kh) * K3_PITCH;
#pragma unroll
      for (int n = 0; n < 4; ++n) {
        int col = n * 16 + ln;
        f32x2 bb = {bp[col], bp[K3_PITCH + col]};  // rows k, k+1 at same column
        acc[n] = wmma_f32_16x16x4(af[kk], bb, acc[n]);
      }
    }

#pragma unroll
    for (int n = 0; n < 4; ++n) {
#pragma unroll
      for (int v = 0; v < 8; ++v) {
        int c = wave * 16 + v + 8 * kh;
        if (c < CH)
          ob[(size_t)c * SEQ + sub * K3_SUB + n * 16 + ln] = acc[n][v];
      }
    }
  }
}

// ---------------------------------------------------------------------------
extern "C" void kernel_launch(void* const* d_in, const int* in_sizes, int n_in,
                              void* d_out, int out_size, void* d_ws, size_t ws_size,
                              hipStream_t stream) {
  (void)in_sizes; (void)n_in; (void)out_size; (void)ws_size;
  const float* x  = (const float*)d_in[0];
  const float* Wq = (const float*)d_in[1];
  const float* bq = (const float*)d_in[2];
  const float* Wk = (const float*)d_in[3];
  const float* bk = (const float*)d_in[4];
  const float* Wv = (const float*)d_in[5];
  const float* bv = (const float*)d_in[6];

  float* G    = (float*)d_ws;                 // [B,96,96]
  float* Mext = G + (size_t)BATCH * CP * CP;  // [B,96,96]
  float* out  = (float*)d_out;                // [B,85,8192]

  (void)hipMemsetAsync(G, 0, (size_t)BATCH * CP * CP * sizeof(float), stream);
  gram_kernel<<<dim3(BATCH, NCHUNK1), 192, 0, stream>>>(x, G);
  attn_kernel<<<dim3(BATCH), 256, 0, stream>>>(G, Wq, bq, Wk, bk, Wv, bv, Mext);
  out_kernel<<<dim3(BATCH, SEQ / K3_CHUNK), 192, 0, stream>>>(x, Mext, out);
}